// GCNII_13907104104746
// MI455X (gfx1250) — compile-verified
//
#include <hip/hip_runtime.h>
#include <cmath>

#define IN_F 64
#define HID  96
#define OUT_F 16
#define NLAYERS 16

typedef __attribute__((ext_vector_type(2))) float v2f;
typedef __attribute__((ext_vector_type(8))) float v8f;

// f32 WMMA: D(16x16) = A(16x4) * B(4x16) + C, wave32, 2 VGPRs for A/B, 8 for C/D.
__device__ __forceinline__ v8f wmma_f32_k4(v2f a, v2f b, v8f c) {
  return __builtin_amdgcn_wmma_f32_16x16x4_f32(false, a, false, b, (short)0, c,
                                               false, false);
}

// ---------------- degree / normalization ----------------
__global__ void k_init_deg(float* __restrict__ deg, int n) {
  int i = blockIdx.x * blockDim.x + threadIdx.x;
  if (i < n) deg[i] = 1.0f;  // self-loop contributes 1 to every degree
}

__global__ void k_count_deg(const int* __restrict__ col, float* __restrict__ deg, int e) {
  int i = blockIdx.x * blockDim.x + threadIdx.x;
  if (i < e) atomicAdd(&deg[col[i]], 1.0f);
}

__global__ void k_to_dis(float* __restrict__ deg, int n) {
  int i = blockIdx.x * blockDim.x + threadIdx.x;
  if (i < n) deg[i] = rsqrtf(deg[i]);  // deg >= 1 always (self-loop)
}

__global__ void k_zero(float* __restrict__ p, int n) {
  int i = blockIdx.x * blockDim.x + threadIdx.x;
  if (i < n) p[i] = 0.0f;
}

// ---------------- lin0: h = relu(x @ w0^T + b0), x[N,64], w0[96,64] ----------------
#define L0_STRIDE 68  // 64 + 4 pad: LDS bank-conflict-free b64 reads
__global__ __launch_bounds__(192) void k_lin0(const float* __restrict__ x,
                                              const float* __restrict__ w0,
                                              const float* __restrict__ b0,
                                              float* __restrict__ out) {
  __shared__ float s[16 * L0_STRIDE];
  const int row0 = blockIdx.x * 16;
  const int tid = threadIdx.x;
  for (int idx = tid; idx < 16 * IN_F; idx += 192) {
    int r = idx >> 6, c = idx & 63;
    s[r * L0_STRIDE + c] = x[(row0 + r) * IN_F + c];
  }
  __syncthreads();
  const int wave = tid >> 5, lane = tid & 31;
  const int j0 = wave * 16, ln = lane & 15, kh = (lane >> 4) * 2;
  v8f acc = {};
#pragma unroll
  for (int k0 = 0; k0 < IN_F; k0 += 4) {
    v2f a, b;
    a.x = s[ln * L0_STRIDE + k0 + kh];
    a.y = s[ln * L0_STRIDE + k0 + kh + 1];
    // B[k][j] = w0[j*IN_F + k]
    b.x = w0[(j0 + ln) * IN_F + k0 + kh];
    b.y = w0[(j0 + ln) * IN_F + k0 + kh + 1];
    acc = wmma_f32_k4(a, b, acc);
  }
  const int mbase = (lane < 16) ? 0 : 8;
  const float bias = b0[j0 + ln];
#pragma unroll
  for (int r = 0; r < 8; ++r) {
    int M = mbase + r;
    out[(row0 + M) * HID + j0 + ln] = fmaxf(acc[r] + bias, 0.0f);
  }
}

// ---------------- SpMM: m[row] += dis[row]*dis[col] * h[col], one wave per edge ----
__global__ __launch_bounds__(256) void k_spmm(const int* __restrict__ rowi,
                                              const int* __restrict__ coli,
                                              const float* __restrict__ dis,
                                              const float* __restrict__ h,
                                              float* __restrict__ m, int e) {
  const int gwave = (blockIdx.x * 256 + threadIdx.x) >> 5;
  if (gwave >= e) return;
  const int lane = threadIdx.x & 31;
  const int r = rowi[gwave];
  const int c = coli[gwave];
  const float v = dis[r] * dis[c];
  const float* hc = h + (size_t)c * HID;
  float* mr = m + (size_t)r * HID;
#pragma unroll
  for (int f = 0; f < HID; f += 32) {
    atomicAdd(&mr[f + lane], v * hc[f + lane]);
  }
}

// -------- combine: s = .9*(m + dis^2*h) + .1*x0 ; out = relu((1-b)*s + b*(s@W)) -----
#define CB_STRIDE 100  // 96 + 4 pad
__global__ __launch_bounds__(192) void k_combine(const float* __restrict__ m,
                                                 const float* __restrict__ x0,
                                                 const float* __restrict__ h,
                                                 const float* __restrict__ dis,
                                                 const float* __restrict__ W,
                                                 float beta,
                                                 float* __restrict__ out) {
  __shared__ float s[16 * CB_STRIDE];
  const int row0 = blockIdx.x * 16;
  const int tid = threadIdx.x;
  for (int idx = tid; idx < 16 * HID; idx += 192) {
    int r = idx / HID, c = idx - r * HID;
    size_t g = (size_t)(row0 + r) * HID + c;
    float d = dis[row0 + r];
    // self-loop folded here: full SpMM result = m + dis^2 * h
    s[r * CB_STRIDE + c] = 0.9f * (m[g] + d * d * h[g]) + 0.1f * x0[g];
  }
  __syncthreads();
  const int wave = tid >> 5, lane = tid & 31;
  const int j0 = wave * 16, ln = lane & 15, kh = (lane >> 4) * 2;
  v8f acc = {};
#pragma unroll
  for (int k0 = 0; k0 < HID; k0 += 4) {
    v2f a, b;
    a.x = s[ln * CB_STRIDE + k0 + kh];
    a.y = s[ln * CB_STRIDE + k0 + kh + 1];
    b.x = W[(k0 + kh) * HID + j0 + ln];
    b.y = W[(k0 + kh + 1) * HID + j0 + ln];
    acc = wmma_f32_k4(a, b, acc);
  }
  const float omb = 1.0f - beta;
  const int mbase = (lane < 16) ? 0 : 8;
#pragma unroll
  for (int r = 0; r < 8; ++r) {
    int M = mbase + r;
    float sv = s[M * CB_STRIDE + j0 + ln];
    // safe to alias out==m: all reads of this strip happened before the barrier
    out[(size_t)(row0 + M) * HID + j0 + ln] = fmaxf(omb * sv + beta * acc[r], 0.0f);
  }
}

// ---------------- lin1: out = h @ w1^T + b1, one wave per 16-row strip -------------
__global__ __launch_bounds__(256) void k_lin1(const float* __restrict__ h,
                                              const float* __restrict__ w1,
                                              const float* __restrict__ b1,
                                              float* __restrict__ out, int nStrips) {
  const int gwave = (blockIdx.x * 256 + threadIdx.x) >> 5;
  if (gwave >= nStrips) return;  // whole-wave exit only (strip == wave granularity)
  const int lane = threadIdx.x & 31;
  const int row0 = gwave * 16;
  const int ln = lane & 15, kh = (lane >> 4) * 2;
  v8f acc = {};
#pragma unroll
  for (int k0 = 0; k0 < HID; k0 += 4) {
    v2f a, b;
    a.x = h[(size_t)(row0 + ln) * HID + k0 + kh];
    a.y = h[(size_t)(row0 + ln) * HID + k0 + kh + 1];
    // B[k][o] = w1[o*HID + k]
    b.x = w1[ln * HID + k0 + kh];
    b.y = w1[ln * HID + k0 + kh + 1];
    acc = wmma_f32_k4(a, b, acc);
  }
  const int mbase = (lane < 16) ? 0 : 8;
  const float bias = b1[ln];
#pragma unroll
  for (int r = 0; r < 8; ++r) {
    int M = mbase + r;
    out[(size_t)(row0 + M) * OUT_F + ln] = acc[r] + bias;
  }
}

extern "C" void kernel_launch(void* const* d_in, const int* in_sizes, int n_in,
                              void* d_out, int out_size, void* d_ws, size_t ws_size,
                              hipStream_t stream) {
  const float* x  = (const float*)d_in[0];
  const int*   ei = (const int*)d_in[1];
  const float* w0 = (const float*)d_in[2];
  const float* b0 = (const float*)d_in[3];
  const float* w1 = (const float*)d_in[4];
  const float* b1 = (const float*)d_in[5];
  const float* cw = (const float*)d_in[6];

  const int N = in_sizes[0] / IN_F;
  const int E = in_sizes[1] / 2;
  const int* rowi = ei;       // edge_index[0]
  const int* coli = ei + E;   // edge_index[1]

  // workspace: dis[N] | X0[N*96] | B0[N*96] | B1[N*96]
  float* dis = (float*)d_ws;
  float* X0  = dis + N;
  float* B0  = X0 + (size_t)N * HID;
  float* B1  = B0 + (size_t)N * HID;

  const int strips = N / 16;  // N = 50000 = 3125 * 16

  k_init_deg<<<(N + 255) / 256, 256, 0, stream>>>(dis, N);
  k_count_deg<<<(E + 255) / 256, 256, 0, stream>>>(coli, dis, E);
  k_to_dis<<<(N + 255) / 256, 256, 0, stream>>>(dis, N);

  k_lin0<<<strips, 192, 0, stream>>>(x, w0, b0, X0);

  const float* hcur = X0;
  float* bufs[2] = {B0, B1};
  const int nfeat = N * HID;
  for (int l = 0; l < NLAYERS; ++l) {
    float* mbuf = bufs[l & 1];  // scratch for SpMM, then aliased as layer output
    k_zero<<<(nfeat + 255) / 256, 256, 0, stream>>>(mbuf, nfeat);
    k_spmm<<<(E + 7) / 8, 256, 0, stream>>>(rowi, coli, dis, hcur, mbuf, E);
    float beta = logf(0.5f / (float)(l + 1) + 1.0f);
    k_combine<<<strips, 192, 0, stream>>>(mbuf, X0, hcur, dis,
                                          cw + (size_t)l * HID * HID, beta, mbuf);
    hcur = mbuf;
  }

  k_lin1<<<(strips * 32 + 255) / 256, 256, 0, stream>>>(hcur, w1, b1,
                                                        (float*)d_out, strips);
}